// DentateGyrus_37580963840254
// MI455X (gfx1250) — compile-verified
//
#include <hip/hip_runtime.h>

// CDNA5 / gfx1250 WMMA vector types
typedef __attribute__((ext_vector_type(2))) float v2f;   // A/B frag of V_WMMA_F32_16X16X4_F32 (2 VGPRs)
typedef __attribute__((ext_vector_type(8))) float v8f;   // 16x16 f32 C/D frag (8 VGPRs)

#define ENTRY_DIM 8192
#define NCELLS    32768
#define ROWS_PER_WAVE  16
#define WAVES_PER_BLOCK 8
#define ROWS_PER_BLOCK (ROWS_PER_WAVE * WAVES_PER_BLOCK)   // 128
#define DT 0.5f

// GEMV (W @ ec)*10 fused with the Izhikevich threshold.
// Reference's top-k pass on a *binary* spike vector is an exact identity
// (threshold is 0 or 1; (s>=thr)*s == s in both cases), so the output is
// just the binary spike vector -> single fused kernel, no reduction.
//
// Per-wave: 16 rows of W. A-matrix (16x4 fp32) ISA layout:
//   lanes 0-15  : VGPR0 = W[row, j+0], VGPR1 = W[row, j+1]   (K=0,1)
//   lanes 16-31 : VGPR0 = W[row, j+2], VGPR1 = W[row, j+3]   (K=2,3)
// B-matrix (4x16 fp32): same K split across lane halves; we replicate the
// ec chunk into all 16 columns (every lane in a half reads the same ec
// address), so every column of the 16x16 accumulator equals the 16-row
// partial dot product. Four accumulators break the WMMA C-dep chain.
__global__ __launch_bounds__(256, 1)
void dg_spike_wmma_kernel(const float* __restrict__ W,
                          const float* __restrict__ ec,
                          const float* __restrict__ v_in,
                          const float* __restrict__ u_in,
                          float* __restrict__ out) {
    const int lane = threadIdx.x & 31;
    const int wave = threadIdx.x >> 5;
    const int row_base = blockIdx.x * ROWS_PER_BLOCK + wave * ROWS_PER_WAVE;

    const int co = (lane >> 4) << 1;   // 0 for lanes 0-15, 2 for lanes 16-31
    const int m  = lane & 15;          // A-matrix row within the 16-row tile

    const float* __restrict__ Arow = W + (size_t)(row_base + m) * ENTRY_DIM + co;
    const float* __restrict__ Bcol = ec + co;

    v8f acc0 = {}, acc1 = {}, acc2 = {}, acc3 = {};

    // 8192 / 16 = 512 iterations, 4 WMMAs each (K=4 per WMMA).
#pragma unroll 2
    for (int j = 0; j < ENTRY_DIM; j += 16) {
        // 8-byte loads; lane pair (L, L+16) covers 16 contiguous bytes of
        // one W row -> clean streaming, lines fully consumed from L0.
        v2f a0 = *(const v2f*)(Arow + j + 0);
        v2f a1 = *(const v2f*)(Arow + j + 4);
        v2f a2 = *(const v2f*)(Arow + j + 8);
        v2f a3 = *(const v2f*)(Arow + j + 12);
        v2f b0 = *(const v2f*)(Bcol + j + 0);    // broadcast within lane half
        v2f b1 = *(const v2f*)(Bcol + j + 4);
        v2f b2 = *(const v2f*)(Bcol + j + 8);
        v2f b3 = *(const v2f*)(Bcol + j + 12);

        acc0 = __builtin_amdgcn_wmma_f32_16x16x4_f32(false, a0, false, b0,
                                                     (short)0, acc0, false, false);
        acc1 = __builtin_amdgcn_wmma_f32_16x16x4_f32(false, a1, false, b1,
                                                     (short)0, acc1, false, false);
        acc2 = __builtin_amdgcn_wmma_f32_16x16x4_f32(false, a2, false, b2,
                                                     (short)0, acc2, false, false);
        acc3 = __builtin_amdgcn_wmma_f32_16x16x4_f32(false, a3, false, b3,
                                                     (short)0, acc3, false, false);
    }

    v8f acc;
#pragma unroll
    for (int r = 0; r < 8; ++r)
        acc[r] = (acc0[r] + acc1[r]) + (acc2[r] + acc3[r]);

    // C/D layout: VGPR r, lanes 0-15 -> (M=r, N=lane); lanes 16-31 -> (M=8+r).
    // All columns identical, so lane 0 holds rows row_base+0..7 in acc[0..7],
    // lane 16 holds rows row_base+8..15.
    if (m == 0) {
        const int rbase = row_base + ((lane >> 4) << 3);
#pragma unroll
        for (int r = 0; r < 8; ++r) {
            const int row = rbase + r;
            const float I  = 10.0f * acc[r];
            const float v  = v_in[row];
            const float u  = u_in[row];
            const float dv = fmaf(0.04f * v, v, fmaf(5.0f, v, 140.0f)) - u + I;
            const float vn = fmaf(dv, DT, v);
            out[row] = (vn >= 30.0f) ? 1.0f : 0.0f;
        }
    }
}

extern "C" void kernel_launch(void* const* d_in, const int* in_sizes, int n_in,
                              void* d_out, int out_size, void* d_ws, size_t ws_size,
                              hipStream_t stream) {
    (void)in_sizes; (void)n_in; (void)d_ws; (void)ws_size; (void)out_size;
    const float* ec = (const float*)d_in[0];   // (8192,)
    const float* W  = (const float*)d_in[1];   // (32768, 8192)
    const float* v  = (const float*)d_in[2];   // membrane_potential
    const float* u  = (const float*)d_in[3];   // recovery_variable
    float* out = (float*)d_out;                // (32768,) binary spikes

    dim3 grid(NCELLS / ROWS_PER_BLOCK);        // 256 blocks
    dim3 block(256);                           // 8 wave32s
    dg_spike_wmma_kernel<<<grid, block, 0, stream>>>(W, ec, v, u, out);
}